// DiscoNetFusion_661424963688
// MI455X (gfx1250) — compile-verified
//
#include <hip/hip_runtime.h>
#include <math.h>

// ---------------------------------------------------------------------------
// DiscoNet fusion for MI455X (gfx1250).
// Only updated[:,0] is consumed by the reference, so we compute target 0 only.
// fp32 everywhere (matches reference numerics); matrix work uses
// V_WMMA_F32_16X16X4_F32 (wave32, 16x16 tiles, K=4 fp32 steps).
// ---------------------------------------------------------------------------

#define B_  2
#define L_  4
#define C_  128
#define H_  96
#define W_  96
#define P_  (H_*W_)          // 9216 pixels
#define INV_RATIO_DS (1.0f/1.6f)

typedef float v2f __attribute__((ext_vector_type(2)));
typedef float v8f __attribute__((ext_vector_type(8)));

// workspace layout (units: floats). total ~14.3M floats (~57 MB)
#define OFF_SEL   0                                   // [B][L][P][C] warped/selected feats
#define OFF_EGO   (OFF_SEL   + B_*L_*P_*C_)           // [B][P][C]    ego feats (f-space)
#define OFF_ROI   (OFF_EGO   + B_*P_*C_)              // [B][L][P]
#define OFF_LOGIT (OFF_ROI   + B_*L_*P_)              // [B][L][P]
#define OFF_FUSED (OFF_LOGIT + B_*L_*P_)              // [B][P][C]  (final pixel order)
#define OFF_STD   (OFF_FUSED + B_*P_*C_)              // [B][C]
#define OFF_CSTAR (OFF_STD   + B_*C_)                 // [B] ints

// ---------------------------------------------------------------------------
// 1) per-channel std (ddof=1) of ego features. std is permutation-invariant,
//    so compute directly on x[b*L+0, c, :, :].
// ---------------------------------------------------------------------------
__global__ void k_std(const float* __restrict__ x, float* __restrict__ ws) {
  int b = blockIdx.x / C_;
  int c = blockIdx.x % C_;
  const float* img = x + ((size_t)(b*L_)*C_ + c) * (size_t)P_;
  float s = 0.f, s2 = 0.f;
  for (int i = threadIdx.x; i < P_; i += blockDim.x) {
    float v = img[i]; s += v; s2 += v*v;
  }
  __shared__ float sh1[256], sh2[256];
  sh1[threadIdx.x] = s; sh2[threadIdx.x] = s2;
  __syncthreads();
  for (int st = 128; st > 0; st >>= 1) {
    if (threadIdx.x < st) { sh1[threadIdx.x] += sh1[threadIdx.x+st];
                            sh2[threadIdx.x] += sh2[threadIdx.x+st]; }
    __syncthreads();
  }
  if (threadIdx.x == 0) {
    float n   = (float)P_;
    float var = (sh2[0] - sh1[0]*sh1[0]/n) / (n - 1.0f);
    ws[OFF_STD + b*C_ + c] = sqrtf(fmaxf(var, 0.0f));
  }
}

__global__ void k_argmax(float* __restrict__ ws) {
  int b = blockIdx.x;
  if (threadIdx.x == 0) {
    const float* s = ws + OFF_STD + b*C_;
    int best = 0; float bv = s[0];
    for (int c = 1; c < C_; ++c) if (s[c] > bv) { bv = s[c]; best = c; }  // first max
    ((int*)ws)[OFF_CSTAR + b] = best;
  }
}

// ---------------------------------------------------------------------------
// 2) affine warp (target 0) + ROI + channel selection.
//    f-space: f[b,n,c,y,x] = x[b*L+n, c, H-1-x, y]  (transpose+flip remap).
// ---------------------------------------------------------------------------
__global__ void k_warp(const float* __restrict__ x, const float* __restrict__ Pm4,
                       float* __restrict__ ws) {
  int gid = blockIdx.x * blockDim.x + threadIdx.x;   // covers B*L*P exactly
  int bn = gid / P_;
  int p  = gid % P_;
  int b  = bn / L_, n = bn % L_;

  // tm[b, n, 0]: rows :2, cols [0,1,3]; translation scaled by 1/(RATIO*DS)
  const float* Pm = Pm4 + ((size_t)(b*L_ + n)*L_ + 0) * 16;
  float r00 = Pm[0], r01 = Pm[1], tx = Pm[3]*INV_RATIO_DS;
  float r10 = Pm[4], r11 = Pm[5], ty = Pm[7]*INV_RATIO_DS;
  // rotation about image center
  const float cx = W_*0.5f, cy = H_*0.5f;
  float t0 = cx - (r00*cx + r01*cy) + tx;
  float t1 = cy - (r10*cx + r11*cy) + ty;
  // theta = inv(n * Mh * inv(n))[:2]  — closed form for affine K
  const float a = 2.0f/(W_-1), d = 2.0f/(H_-1);
  float k00 = r00,        k01 = r01*a/d;
  float k02 = k00 + k01 + a*t0 - 1.0f;
  float k10 = r10*d/a,    k11 = r11;
  float k12 = k10 + k11 + d*t1 - 1.0f;
  float idet = 1.0f / (k00*k11 - k01*k10);
  float i00 =  k11*idet, i01 = -k01*idet;
  float i10 = -k10*idet, i11 =  k00*idet;
  float it0 = -(i00*k02 + i01*k12);
  float it1 = -(i10*k02 + i11*k12);

  int y = p / W_, xx = p % W_;
  float gx = a*xx - 1.0f, gy = d*y - 1.0f;
  float u = i00*gx + i01*gy + it0;
  float v = i10*gx + i11*gy + it1;
  float sx = (u + 1.0f)*0.5f*(W_-1);
  float sy = (v + 1.0f)*0.5f*(H_-1);
  float x0f = floorf(sx), y0f = floorf(sy);
  float wx = sx - x0f,    wy = sy - y0f;
  int x0 = (int)x0f, y0 = (int)y0f, x1 = x0+1, y1 = y0+1;
  float v00 = (x0>=0 && x0<W_ && y0>=0 && y0<H_) ? 1.f : 0.f;
  float v10 = (x1>=0 && x1<W_ && y0>=0 && y0<H_) ? 1.f : 0.f;
  float v01 = (x0>=0 && x0<W_ && y1>=0 && y1<H_) ? 1.f : 0.f;
  float v11 = (x1>=0 && x1<W_ && y1>=0 && y1<H_) ? 1.f : 0.f;
  int x0c = x0<0?0:(x0>W_-1?W_-1:x0);
  int x1c = x1<0?0:(x1>W_-1?W_-1:x1);
  int y0c = y0<0?0:(y0>H_-1?H_-1:y0);
  int y1c = y1<0?0:(y1>H_-1?H_-1:y1);
  float w00 = (1.f-wx)*(1.f-wy), w10 = wx*(1.f-wy);
  float w01 = (1.f-wx)*wy,       w11 = wx*wy;
  float roi = v00*w00 + v10*w10 + v01*w01 + v11*w11;
  ws[OFF_ROI + bn*P_ + p] = roi;

  int  cstar = ((const int*)ws)[OFF_CSTAR + b];
  bool first = (b == 0);

  // tap offsets into x for f(row=yc, col=xc) = x[..., H-1-xc, yc]
  int o00 = (H_-1-x0c)*W_ + y0c;
  int o10 = (H_-1-x1c)*W_ + y0c;
  int o01 = (H_-1-x0c)*W_ + y1c;
  int o11 = (H_-1-x1c)*W_ + y1c;
  int oe  = (H_-1-xx)*W_ + y;
  float m00 = v00*w00, m10 = v10*w10, m01 = v01*w01, m11 = v11*w11;

  const size_t baseN = (size_t)bn       * C_ * P_;
  const size_t baseE = (size_t)(b*L_)   * C_ * P_;
  float* sel = ws + OFF_SEL + ((size_t)bn*P_ + p) * C_;
  float* ego = ws + OFF_EGO + ((size_t)b*P_ + p) * C_;
  for (int c = 0; c < C_; ++c) {
    const float* img = x + baseN + (size_t)c * P_;
    float nv = m00*img[o00] + m10*img[o10] + m01*img[o01] + m11*img[o11];
    float ev = x[baseE + (size_t)c*P_ + oe];
    sel[c] = (first || c == cstar) ? nv : ev;
    if (n == 0) ego[c] = ev;
  }
}

// ---------------------------------------------------------------------------
// 3) pixel-wise MLP 256->128->32->8->1.  Layer 1 via V_WMMA_F32_16X16X4_F32:
//    one wave per 16 pixels, M=16 (pixels), K=256, N=128 (8 tiles).
//    A frag: lane half selects K pair (K0/K1 vs K2/K3), row = lane%16.
//    C/D:    vgpr r holds M=r (lanes 0-15) / M=8+r (lanes 16-31), N=lane%16.
// ---------------------------------------------------------------------------
__global__ void k_mlp(float* __restrict__ ws,
                      const float* __restrict__ w1, const float* __restrict__ b1,
                      const float* __restrict__ s1, const float* __restrict__ t1,
                      const float* __restrict__ w2, const float* __restrict__ b2,
                      const float* __restrict__ s2, const float* __restrict__ t2,
                      const float* __restrict__ w3, const float* __restrict__ b3,
                      const float* __restrict__ s3, const float* __restrict__ t3,
                      const float* __restrict__ w4, const float* __restrict__ b4) {
  int tile = blockIdx.x;                  // B*L*(P/16) tiles, 1 wave each
  int bn   = tile / (P_/16);
  int p0   = (tile % (P_/16)) * 16;
  int b    = bn / L_;
  int lane = threadIdx.x, half = lane >> 4, lm = lane & 15;

  const float* sel = ws + OFF_SEL + ((size_t)bn*P_ + p0) * C_;
  const float* ego = ws + OFF_EGO + ((size_t)b*P_ + p0) * C_;

  v8f acc[8];
  const v8f vz = {0.f,0.f,0.f,0.f,0.f,0.f,0.f,0.f};
#pragma unroll
  for (int j = 0; j < 8; ++j) acc[j] = vz;

  for (int k0 = 0; k0 < 256; k0 += 4) {     // K loop: concat(sel, ego)
    int kk = k0 + 2*half;
    v2f A;
    if (k0 < 128) {
      A.x = sel[(size_t)lm*C_ + kk];
      A.y = sel[(size_t)lm*C_ + kk + 1];
    } else {
      A.x = ego[(size_t)lm*C_ + (kk - 128)];
      A.y = ego[(size_t)lm*C_ + (kk - 127)];
    }
#pragma unroll
    for (int j = 0; j < 8; ++j) {           // N tiles; B[k][n] = w1[n_glob][k]
      int o = j*16 + lm;
      v2f Bf;
      Bf.x = w1[(size_t)o*256 + kk];
      Bf.y = w1[(size_t)o*256 + kk + 1];
      acc[j] = __builtin_amdgcn_wmma_f32_16x16x4_f32(
          false, A, false, Bf, (short)0, acc[j], false, false);
    }
  }

  __shared__ float lds1[16*C_];             // h1: 16 px x 128 ch
  __shared__ float lds2[16*32];             // h2: 16 px x 32 ch
#pragma unroll
  for (int j = 0; j < 8; ++j) {
    int N = j*16 + lm;
    float bb = b1[N], ss = s1[N], tt = t1[N];
#pragma unroll
    for (int r = 0; r < 8; ++r) {
      int m = r + 8*half;
      lds1[m*C_ + N] = fmaxf((acc[j][r] + bb)*ss + tt, 0.0f);
    }
  }
  __syncthreads();

  { // layer 2: 128->32, 2 lanes per pixel, 16 out-channels each
    int m = lane >> 1, g = lane & 1;
    for (int oj = 0; oj < 16; ++oj) {
      int o = g*16 + oj;
      float a2 = b2[o];
      for (int k = 0; k < C_; ++k) a2 += lds1[m*C_ + k] * w2[o*C_ + k];
      lds2[m*32 + o] = fmaxf(a2*s2[o] + t2[o], 0.0f);
    }
  }
  __syncthreads();

  if (lane < 16) {  // layers 3+4 per pixel
    int m = lane;
    float h3[8];
    for (int o = 0; o < 8; ++o) {
      float a3 = b3[o];
      for (int k = 0; k < 32; ++k) a3 += lds2[m*32 + k] * w3[o*32 + k];
      h3[o] = fmaxf(a3*s3[o] + t3[o], 0.0f);
    }
    float lg = b4[0];
    for (int o = 0; o < 8; ++o) lg += h3[o] * w4[o];
    ws[OFF_LOGIT + bn*P_ + p0 + m] = fmaxf(lg, 0.0f);
  }
}

// ---------------------------------------------------------------------------
// 4) roi-masked softmax over agents + weighted fuse; fold the inverse
//    flip/transpose into the store: out pixel (y,x) -> final (W-1-x, y).
// ---------------------------------------------------------------------------
__global__ void k_fuse(float* __restrict__ ws) {
  int bp = blockIdx.x;            // B*P blocks, 128 threads = channels
  int b  = bp / P_;
  int p  = bp % P_;
  float lg[4], rv[4];
#pragma unroll
  for (int n = 0; n < 4; ++n) {
    lg[n] = ws[OFF_LOGIT + (b*L_ + n)*P_ + p];
    rv[n] = ws[OFF_ROI   + (b*L_ + n)*P_ + p];
  }
  float mx = -3.0e38f;
#pragma unroll
  for (int n = 0; n < 4; ++n) if (rv[n] != 0.f) mx = fmaxf(mx, lg[n]);
  float e[4], ssum = 0.f;
#pragma unroll
  for (int n = 0; n < 4; ++n) {
    e[n] = (rv[n] != 0.f) ? __expf(lg[n] - mx) : 0.f;
    ssum += e[n];
  }
  float inv = 1.0f / ssum;        // agent 0 has roi==1 -> ssum > 0
  int c = threadIdx.x;
  float acc = 0.f;
#pragma unroll
  for (int n = 0; n < 4; ++n)
    acc += (e[n]*inv) * rv[n] *
           ws[OFF_SEL + ((size_t)(b*L_ + n)*P_ + p)*C_ + c];
  int y = p / W_, xx = p % W_;
  int q = (W_-1-xx)*W_ + y;       // inverse flip/transpose
  ws[OFF_FUSED + ((size_t)b*P_ + q)*C_ + c] = acc;
}

// ---------------------------------------------------------------------------
// 5) final per-pixel 128x128 matmul (+bias) via WMMA, stores (B,H,W,C).
// ---------------------------------------------------------------------------
__global__ void k_out(const float* __restrict__ ws, const float* __restrict__ mw,
                      const float* __restrict__ mb, float* __restrict__ out) {
  int tile = blockIdx.x;
  int b    = tile / (P_/16);
  int q0   = (tile % (P_/16)) * 16;
  int lane = threadIdx.x, half = lane >> 4, lm = lane & 15;
  const float* A = ws + OFF_FUSED + ((size_t)b*P_ + q0) * C_;

  v8f acc[8];
  const v8f vz = {0.f,0.f,0.f,0.f,0.f,0.f,0.f,0.f};
#pragma unroll
  for (int j = 0; j < 8; ++j) acc[j] = vz;

  for (int k0 = 0; k0 < C_; k0 += 4) {
    int kk = k0 + 2*half;
    v2f Af;
    Af.x = A[(size_t)lm*C_ + kk];
    Af.y = A[(size_t)lm*C_ + kk + 1];
#pragma unroll
    for (int j = 0; j < 8; ++j) {
      int o = j*16 + lm;
      v2f Bf;
      Bf.x = mw[(size_t)o*C_ + kk];       // out[o] = sum_c fused[c]*mlp_w[o][c]
      Bf.y = mw[(size_t)o*C_ + kk + 1];
      acc[j] = __builtin_amdgcn_wmma_f32_16x16x4_f32(
          false, Af, false, Bf, (short)0, acc[j], false, false);
    }
  }
#pragma unroll
  for (int j = 0; j < 8; ++j) {
    int N = j*16 + lm;
    float bias = mb[N];
#pragma unroll
    for (int r = 0; r < 8; ++r) {
      int m = r + 8*half;
      out[((size_t)(b*P_ + q0 + m))*C_ + N] = acc[j][r] + bias;
    }
  }
}

// ---------------------------------------------------------------------------
extern "C" void kernel_launch(void* const* d_in, const int* in_sizes, int n_in,
                              void* d_out, int out_size, void* d_ws, size_t ws_size,
                              hipStream_t stream) {
  const float* x  = (const float*)d_in[0];
  // d_in[1] = record_len (unused: always L)
  const float* Pm = (const float*)d_in[2];
  const float* w1 = (const float*)d_in[3];
  const float* b1 = (const float*)d_in[4];
  const float* s1 = (const float*)d_in[5];
  const float* t1 = (const float*)d_in[6];
  const float* w2 = (const float*)d_in[7];
  const float* b2 = (const float*)d_in[8];
  const float* s2 = (const float*)d_in[9];
  const float* t2 = (const float*)d_in[10];
  const float* w3 = (const float*)d_in[11];
  const float* b3 = (const float*)d_in[12];
  const float* s3 = (const float*)d_in[13];
  const float* t3 = (const float*)d_in[14];
  const float* w4 = (const float*)d_in[15];
  const float* b4 = (const float*)d_in[16];
  const float* mw = (const float*)d_in[17];
  const float* mb = (const float*)d_in[18];
  float* ws  = (float*)d_ws;
  float* out = (float*)d_out;

  k_std   <<<B_*C_,            256, 0, stream>>>(x, ws);
  k_argmax<<<B_,                32, 0, stream>>>(ws);
  k_warp  <<<(B_*L_*P_)/256,   256, 0, stream>>>(x, Pm, ws);
  k_mlp   <<<B_*L_*(P_/16),     32, 0, stream>>>(ws, w1,b1,s1,t1, w2,b2,s2,t2,
                                                 w3,b3,s3,t3, w4,b4);
  k_fuse  <<<B_*P_,            C_, 0, stream>>>(ws);
  k_out   <<<B_*(P_/16),        32, 0, stream>>>(ws, mw, mb, out);
}